// TensorAccumulator_24661702213844
// MI455X (gfx1250) — compile-verified
//
#include <hip/hip_runtime.h>
#include <stdint.h>

#define BA       8
#define DB_DIM   64
#define HW       65536
#define DB_SIZE  1000000
#define NSEL     10000

// ---------------------------------------------------------------------------
// Kernel 1: priority-table init (one int per db column, -1 = never written)
// ---------------------------------------------------------------------------
__global__ void taccum_init_ws(int* __restrict__ ws, int n) {
    int i = blockIdx.x * blockDim.x + threadIdx.x;
    if (i < n) ws[i] = -1;
}

// ---------------------------------------------------------------------------
// Kernel 2: bulk copy db -> out (256 MB) on the CDNA5 async data-mover path.
// Tile = 4 back-to-back async b128 loads per lane (2 KB per wave in flight),
// double-buffered by tile parity so ONE s_wait_asynccnt per tile suffices:
//   - tile T+1 loads write the opposite LDS half from tile T's pending
//     stores -> safe to issue before any wait;
//   - the mid-tile wait retires T's stores + T+1's loads before T+1 stores;
//   - half p is reused at tile T+2, whose loads issue only after that wait.
// Non-temporal hints keep this 512 MB stream from evicting the scatter's
// L2-resident embed working set.
// ---------------------------------------------------------------------------
#define CPY_BLOCK  256
#define CPY_UNROLL 4
__global__ __launch_bounds__(CPY_BLOCK)
void taccum_copy_db(const float4* __restrict__ src, float4* __restrict__ dst,
                    long long n4) {
    __shared__ char smem[CPY_BLOCK * 16 * CPY_UNROLL * 2];   // 32 KB
    unsigned ldsbase = (unsigned)(uintptr_t)&smem[threadIdx.x * 16];

    long long tile_stride = (long long)gridDim.x * CPY_BLOCK * CPY_UNROLL;
    long long base = (long long)blockIdx.x * CPY_BLOCK * CPY_UNROLL + threadIdx.x;
    int parity = 0;

    for (; base < n4; base += tile_stride) {
        unsigned half = ldsbase + (unsigned)parity * (CPY_BLOCK * 16 * CPY_UNROLL);
        // issue CPY_UNROLL coalesced async loads back-to-back
        #pragma unroll
        for (int k = 0; k < CPY_UNROLL; ++k) {
            long long i = base + (long long)k * CPY_BLOCK;
            if (i < n4) {
                const float4* s = src + i;
                unsigned      l = half + (unsigned)k * (CPY_BLOCK * 16);
                asm volatile("global_load_async_to_lds_b128 %0, %1, off th:TH_LOAD_NT"
                             :: "v"(l), "v"(s) : "memory");
            }
        }
        // single wait per tile: retires this tile's loads AND last tile's stores
        asm volatile("s_wait_asynccnt 0x0" ::: "memory");
        #pragma unroll
        for (int k = 0; k < CPY_UNROLL; ++k) {
            long long i = base + (long long)k * CPY_BLOCK;
            if (i < n4) {
                float4*  d = dst + i;
                unsigned l = half + (unsigned)k * (CPY_BLOCK * 16);
                asm volatile("global_store_async_from_lds_b128 %0, %1, off th:TH_STORE_NT"
                             :: "v"(d), "v"(l) : "memory");
            }
        }
        parity ^= 1;
    }
    asm volatile("s_wait_asynccnt 0x0" ::: "memory");
}

// ---------------------------------------------------------------------------
// Kernel 3: priority pass. e = b*NSEL + j encodes the reference's sequential
// scan order; atomicMax gives deterministic "last (b,j) wins".
// ---------------------------------------------------------------------------
__global__ void taccum_priority(const int* __restrict__ idx_replace,
                                int* __restrict__ ws) {
    int e = blockIdx.x * blockDim.x + threadIdx.x;
    if (e < BA * NSEL) {
        atomicMax(&ws[idx_replace[e]], e);
    }
}

// ---------------------------------------------------------------------------
// Kernel 4: one 64-thread block per update. blockIdx-based indexing makes
// idx_replace[e], ws[r], idx[e] wave-uniform -> scalar (s_load) path; losing
// blocks retire after two dependent scalar loads + uniform branch. Winners move their
// 64-float column (inherently strided; embed slices are L2-resident).
// ---------------------------------------------------------------------------
__global__ __launch_bounds__(DB_DIM)
void taccum_scatter(const float* __restrict__ embed,
                    const int* __restrict__ idx,
                    const int* __restrict__ idx_replace,
                    const int* __restrict__ ws,
                    float* __restrict__ out) {
    int e = blockIdx.x;                  // update id in scan order
    int r = idx_replace[e];              // uniform -> s_load
    if (ws[r] != e) return;              // not the last writer for column r
    int d   = threadIdx.x;               // channel
    int b   = e / NSEL;
    int pix = idx[e];                    // uniform -> s_load
    out[(long long)d * DB_SIZE + r] =
        embed[(long long)(b * DB_DIM + d) * HW + pix];
}

// ---------------------------------------------------------------------------
extern "C" void kernel_launch(void* const* d_in, const int* in_sizes, int n_in,
                              void* d_out, int out_size, void* d_ws, size_t ws_size,
                              hipStream_t stream) {
    const float* embed   = (const float*)d_in[0];  // (8, 64, 256, 256) f32
    const float* db      = (const float*)d_in[1];  // (64, 1000000) f32
    const int*   idx     = (const int*)d_in[2];    // (8, 10000) i32
    const int*   idx_rep = (const int*)d_in[3];    // (8, 10000) i32
    float*       out     = (float*)d_out;          // (64, 1000000) f32
    int*         ws      = (int*)d_ws;             // DB_SIZE ints (4 MB)

    // 1) priority table = -1 (every call; harness doesn't re-poison d_ws)
    taccum_init_ws<<<(DB_SIZE + 255) / 256, 256, 0, stream>>>(ws, DB_SIZE);

    // 2) stream db -> out. n4 = 16M chunks = 15625 tiles of 1024;
    //    3125 blocks x 5 tiles each -> exact cover, no tail.
    long long n4 = (long long)DB_DIM * DB_SIZE / 4;
    taccum_copy_db<<<3125, CPY_BLOCK, 0, stream>>>((const float4*)db,
                                                   (float4*)out, n4);

    // 3) deterministic last-writer resolution
    taccum_priority<<<(BA * NSEL + 255) / 256, 256, 0, stream>>>(idx_rep, ws);

    // 4) apply winning column updates (80000 blocks x 64 threads)
    taccum_scatter<<<BA * NSEL, DB_DIM, 0, stream>>>(embed, idx, idx_rep, ws, out);
}